// _ScaledDotProductAttention_39410619908734
// MI455X (gfx1250) — compile-verified
//
#include <hip/hip_runtime.h>
#include <hip/hip_bf16.h>
#include <math.h>

typedef __attribute__((ext_vector_type(2))) float v2f;
typedef __attribute__((ext_vector_type(8))) float v8f;

// D = A(16x4 f32) * B(4x16 f32) + C(16x16 f32), wave32
#define WMMA_F32(a, b, c) \
  __builtin_amdgcn_wmma_f32_16x16x4_f32(false, (a), false, (b), (short)0, (c), false, false)

static constexpr int B_ = 2, H_ = 16, S_ = 1536, D_ = 64;
static constexpr int BH_ = B_ * H_;          // 32
static constexpr int QT_ = S_ / 16;          // 96 query tiles per (b,h)

// ---------------------------------------------------------------------------
// Kernel 1: c = layernorm(ctx) ; cb = c @ bilinear  (both into workspace)
// grid = BH*S/16 blocks of 128 threads (4 waves). Each block: 16 rows.
// ---------------------------------------------------------------------------
__global__ void ln_cb_kernel(const float* __restrict__ ctx,
                             const float* __restrict__ bilinear,
                             const float* __restrict__ lnw,
                             const float* __restrict__ lnb,
                             float* __restrict__ cws,
                             float* __restrict__ cbws) {
  __shared__ float cs[16][64];   // 4 KB: layernormed tile
  __shared__ float bl[64][64];   // 16 KB: bilinear matrix (L2-resident, tiny)
  const int tid = threadIdx.x;
  const long rowbase = (long)blockIdx.x * 16;   // row index in [BH*S] space

  for (int i = tid; i < 64 * 64; i += 128) bl[i >> 6][i & 63] = bilinear[i];

  // LayerNorm: 8 lanes per row, 8 elems per lane (groups stay inside a wave32)
  {
    const int r = tid >> 3;
    const int g = tid & 7;
    const float* x = ctx + (rowbase + r) * D_;
    float vals[8];
    float s = 0.f, ss = 0.f;
    #pragma unroll
    for (int i = 0; i < 8; ++i) {
      float v = x[g * 8 + i];
      vals[i] = v; s += v; ss += v * v;
    }
    #pragma unroll
    for (int o = 4; o; o >>= 1) {
      s  += __shfl_xor(s,  o, 8);
      ss += __shfl_xor(ss, o, 8);
    }
    const float mu   = s * (1.f / 64.f);
    const float var  = ss * (1.f / 64.f) - mu * mu;
    const float rstd = rsqrtf(var + 1e-5f);
    #pragma unroll
    for (int i = 0; i < 8; ++i) {
      const int d = g * 8 + i;
      const float cv = (vals[i] - mu) * rstd * lnw[d] + lnb[d];
      cs[r][d] = cv;
      cws[(rowbase + r) * D_ + d] = cv;
    }
  }
  __syncthreads();

  // cb tile via WMMA f32 16x16x4: wave w -> output cols [16w, 16w+16)
  const int wave = tid >> 5, lane = tid & 31;
  const int half = lane >> 4, mn = lane & 15;
  const int col0 = wave * 16;
  v8f acc = {};
  #pragma unroll
  for (int kk = 0; kk < 64; kk += 4) {
    v2f a, b;
    a.x = cs[mn][kk + 2 * half];
    a.y = cs[mn][kk + 2 * half + 1];
    b.x = bl[kk + 2 * half][col0 + mn];
    b.y = bl[kk + 2 * half + 1][col0 + mn];
    acc = WMMA_F32(a, b, acc);
  }
  #pragma unroll
  for (int i = 0; i < 8; ++i) {
    const int m = i + 8 * half;
    cbws[(rowbase + m) * D_ + col0 + mn] = acc[i];
  }
}

// ---------------------------------------------------------------------------
// Kernel 2: fused scores / mask / softmax / bilinear-bias / l2norm / out-GEMM.
// One block per (bh, 16-query-row tile): 128 threads, ~200KB dynamic LDS
// (CDNA5 WGP has 320KB LDS). bb never touches HBM.
// ---------------------------------------------------------------------------
__global__ void attn_kernel(const float* __restrict__ q,
                            const float* __restrict__ k,
                            const float* __restrict__ v,
                            const float* __restrict__ prev,
                            const float* __restrict__ cws,
                            const float* __restrict__ cbws,
                            const float* __restrict__ bias_scale_p,
                            float* __restrict__ out,
                            float* __restrict__ attn_out,
                            float* __restrict__ scores_out) {
  extern __shared__ float smem[];
  float* sc  = smem;                   // 16*1536: scores row-block -> probs -> attn
  float* bb  = smem + 16 * S_;         // 16*1536: bilinear bias row-block
  float* qt  = smem + 32 * S_;         // 16*64
  float* cbt = qt + 16 * 64;           // 16*64

  const int tid  = threadIdx.x;
  const int wave = tid >> 5, lane = tid & 31;
  const int half = lane >> 4, mn = lane & 15;

  const int bh   = blockIdx.x / QT_;
  const int row0 = (blockIdx.x % QT_) * 16;
  const long rbase = (long)bh * S_ + row0;

  const float scale  = 0.125f;                // D^-0.5, D = 64
  const float bscale = bias_scale_p[0];

  for (int i = tid; i < 16 * 64; i += 128) {
    const int r = i >> 6, d = i & 63;
    qt[i]  = q[(rbase + r) * D_ + d];
    cbt[i] = cbws[(rbase + r) * D_ + d];
  }
  __syncthreads();

  const float* kbh   = k    + (long)bh * D_ * S_;          // [D,S]
  const float* cbh   = cws  + (long)bh * S_ * D_;          // [S,D]
  const float* prevp = prev + rbase * S_;
  float* scout = scores_out + rbase * S_;
  float* aout  = attn_out   + rbase * S_;

  // ---- Phase 1: scores & bb tiles (WMMA f32), scores -> HBM, both -> LDS ----
  for (int j = wave; j < QT_; j += 4) {
    const int col0 = j * 16;
    if (j + 4 < QT_) {  // prefetch next k panel column (global_prefetch_b8)
      __builtin_prefetch(&kbh[(long)(2 * half) * S_ + (j + 4) * 16 + mn], 0, 1);
    }
    v8f accS = {};
    v8f accB = {};
    #pragma unroll
    for (int kk = 0; kk < 64; kk += 4) {
      v2f a, b;
      a.x = qt[mn * 64 + kk + 2 * half];
      a.y = qt[mn * 64 + kk + 2 * half + 1];
      b.x = kbh[(long)(kk + 2 * half) * S_ + col0 + mn];
      b.y = kbh[(long)(kk + 2 * half + 1) * S_ + col0 + mn];
      accS = WMMA_F32(a, b, accS);
      v2f a2, b2;                       // bb = cb @ c^T : B[e][s] = c[s][e]
      a2.x = cbt[mn * 64 + kk + 2 * half];
      a2.y = cbt[mn * 64 + kk + 2 * half + 1];
      b2.x = cbh[(long)(col0 + mn) * D_ + kk + 2 * half];
      b2.y = cbh[(long)(col0 + mn) * D_ + kk + 2 * half + 1];
      accB = WMMA_F32(a2, b2, accB);
    }
    #pragma unroll
    for (int i = 0; i < 8; ++i) {
      const int m = i + 8 * half;       // C/D layout: VGPR i -> rows i, i+8
      const int col = col0 + mn;
      float sv = accS[i] * scale + prevp[(long)m * S_ + col];
      if (col > row0 + m) sv = -__builtin_inff();   // causal mask
      sc[m * S_ + col] = sv;
      scout[(long)m * S_ + col] = sv;
      bb[m * S_ + col] = accB[i];
    }
  }
  __syncthreads();

  // ---- Phase 2: per-row softmax + l2norm(bb) + l2norm(p + s*bbn) ----
  for (int r = wave; r < 16; r += 4) {
    float* srow = sc + r * S_;
    float* brow = bb + r * S_;
    float mx = -__builtin_inff();
    for (int c = lane; c < S_; c += 32) mx = fmaxf(mx, srow[c]);
    #pragma unroll
    for (int o = 16; o; o >>= 1) mx = fmaxf(mx, __shfl_xor(mx, o, 32));

    float se = 0.f, sb2 = 0.f;
    for (int c = lane; c < S_; c += 32) {
      const float e = expf(srow[c] - mx);
      srow[c] = e;
      se += e;
      const float bv = brow[c];
      sb2 += bv * bv;
    }
    #pragma unroll
    for (int o = 16; o; o >>= 1) {
      se  += __shfl_xor(se,  o, 32);
      sb2 += __shfl_xor(sb2, o, 32);
    }
    const float inv_se = 1.f / se;
    const float bmul   = bscale / fmaxf(sqrtf(sb2), 1e-12f);

    float st2 = 0.f;
    for (int c = lane; c < S_; c += 32) {
      const float t = srow[c] * inv_se + brow[c] * bmul;
      srow[c] = t;
      st2 += t * t;
    }
    #pragma unroll
    for (int o = 16; o; o >>= 1) st2 += __shfl_xor(st2, o, 32);
    const float an = 1.f / fmaxf(sqrtf(st2), 1e-12f);

    for (int c = lane; c < S_; c += 32) {
      const float t = srow[c] * an;
      srow[c] = t;
      aout[(long)r * S_ + c] = t;
    }
  }
  __syncthreads();

  // ---- Phase 3: out = attn(16x1536) @ v(1536x64) via WMMA f32 ----
  const float* vbh = v + (long)bh * S_ * D_;
  const int col0 = wave * 16;
  v8f accO = {};
  for (int kk = 0; kk < S_; kk += 4) {
    v2f a, b;
    a.x = sc[mn * S_ + kk + 2 * half];
    a.y = sc[mn * S_ + kk + 2 * half + 1];
    b.x = vbh[(long)(kk + 2 * half) * D_ + col0 + mn];
    b.y = vbh[(long)(kk + 2 * half + 1) * D_ + col0 + mn];
    accO = WMMA_F32(a, b, accO);
  }
  #pragma unroll
  for (int i = 0; i < 8; ++i) {
    const int m = i + 8 * half;
    out[(rbase + m) * D_ + col0 + mn] = accO[i];
  }
}

// ---------------------------------------------------------------------------
extern "C" void kernel_launch(void* const* d_in, const int* in_sizes, int n_in,
                              void* d_out, int out_size, void* d_ws, size_t ws_size,
                              hipStream_t stream) {
  const float* q    = (const float*)d_in[0];
  const float* k    = (const float*)d_in[1];
  const float* v    = (const float*)d_in[2];
  const float* ctx  = (const float*)d_in[3];
  const float* prev = (const float*)d_in[4];
  // d_in[5] = attn_mask (causal triu, recomputed analytically in-kernel)
  const float* bilinear = (const float*)d_in[6];
  const float* bias_scale = (const float*)d_in[7];
  const float* ln_w = (const float*)d_in[8];
  const float* ln_b = (const float*)d_in[9];

  float* out_p    = (float*)d_out;                              // [B,H,S,D]
  float* attn_p   = out_p  + (size_t)BH_ * S_ * D_;             // [B,H,S,S]
  float* scores_p = attn_p + (size_t)BH_ * S_ * S_;             // [B,H,S,S]

  float* c_ws  = (float*)d_ws;                                  // [BH*S, D]
  float* cb_ws = c_ws + (size_t)BH_ * S_ * D_;                  // [BH*S, D]

  const int blocks = BH_ * QT_;   // 3072
  ln_cb_kernel<<<blocks, 128, 0, stream>>>(ctx, bilinear, ln_w, ln_b, c_ws, cb_ws);

  const size_t smem_bytes = (size_t)(2 * 16 * S_ + 2 * 16 * 64) * sizeof(float); // ~200 KB
  attn_kernel<<<blocks, 128, smem_bytes, stream>>>(q, k, v, prev, c_ws, cb_ws,
                                                   bias_scale, out_p, attn_p, scores_p);
}